// SequenceRouter_71442486001883
// MI455X (gfx1250) — compile-verified
//
#include <hip/hip_runtime.h>
#include <hip/hip_bf16.h>
#include <cstddef>

typedef float v2f __attribute__((ext_vector_type(2)));
typedef float v8f __attribute__((ext_vector_type(8)));

#define BDIM 4
#define TDIM 256
#define BT 1024
#define WINDOW 5
#define LPAD 2
#define IN_CAPS 160   // 32 caps * 5 window
#define DIN 8         // input capsule dim for both routing layers

// ---------------------------------------------------------------------------
// block reduction helper (sum) over nthreads (power of 2)
// ---------------------------------------------------------------------------
__device__ __forceinline__ float block_reduce_sum(float v, float* lds, int tid, int nthreads) {
    lds[tid] = v;
    __syncthreads();
    for (int s = nthreads >> 1; s > 0; s >>= 1) {
        if (tid < s) lds[tid] += lds[tid + s];
        __syncthreads();
    }
    float r = lds[0];
    __syncthreads();
    return r;
}

// ---------------------------------------------------------------------------
// 1) primary squash (per 8-dim capsule) + input LayerNorm over 256 features
//    grid: BT blocks of 256 threads; thread t <-> (cap n = t/8, dim e = t%8)
// ---------------------------------------------------------------------------
__global__ __launch_bounds__(256)
void preprocess_kernel(const float* __restrict__ emb, const float* __restrict__ g,
                       const float* __restrict__ b, float* __restrict__ x0) {
    const int bt = blockIdx.x;
    const int t  = threadIdx.x;
    float val = emb[(size_t)bt * 256 + t];
    // per-capsule (8 lanes, aligned within wave32) squared-norm via butterflies
    float sq = val * val;
    sq += __shfl_xor(sq, 1);
    sq += __shfl_xor(sq, 2);
    sq += __shfl_xor(sq, 4);
    float sval = (sq / (1.0f + sq)) * val * rsqrtf(sq + 1e-7f);
    __shared__ float red[256];
    float mean = block_reduce_sum(sval, red, t, 256) * (1.0f / 256.0f);
    float d = sval - mean;
    float var = block_reduce_sum(d * d, red, t, 256) * (1.0f / 256.0f);
    x0[(size_t)bt * 256 + t] = g[t] * d * rsqrtf(var + 1e-3f) + b[t];
}

// ---------------------------------------------------------------------------
// 2) sliding-window gather: xw[bt, w*32+n, e] = src[b, t+w-2, n, e] (0 if OOB)
//    reference ordering: i = w*N + n (concat over w along capsule axis)
// ---------------------------------------------------------------------------
__global__ __launch_bounds__(256)
void gather_kernel(const float* __restrict__ src, float* __restrict__ xw) {
    const int bt = blockIdx.x;
    const int t  = bt % TDIM;
    const int tid = threadIdx.x;        // n*8+e
    #pragma unroll
    for (int w = 0; w < WINDOW; ++w) {
        const int tt = t + w - LPAD;
        float val = (tt >= 0 && tt < TDIM) ? src[(size_t)(bt + (w - LPAD)) * 256 + tid] : 0.0f;
        xw[(size_t)bt * (IN_CAPS * DIN) + w * 256 + tid] = val;
    }
}

// ---------------------------------------------------------------------------
// 3) zero fill (routing logits)
// ---------------------------------------------------------------------------
__global__ void zero_kernel(float* __restrict__ p, int count) {
    int i = blockIdx.x * blockDim.x + threadIdx.x;
    if (i < count) p[i] = 0.0f;
}

// ---------------------------------------------------------------------------
// 4) routing softmax over output caps: c[bt,i,:] = softmax(blog[bt,i,:]+mask)
//    (mask re-applied fresh each iter: -1e9 vs -k*1e9 are numerically identical
//     after max-subtraction, both underflow to 0)
// ---------------------------------------------------------------------------
__global__ void softmax_kernel(const float* __restrict__ blog, float* __restrict__ c,
                               int On, int mask_first, int total) {
    int idx = blockIdx.x * blockDim.x + threadIdx.x;   // bt*In + i
    if (idx >= total) return;
    const float* row = blog + (size_t)idx * On;
    float* crow = c + (size_t)idx * On;
    float mx = -3.4e38f;
    for (int o = 0; o < On; ++o) {
        float l = (mask_first && o == 0) ? -1e9f : row[o];
        mx = fmaxf(mx, l);
    }
    float sum = 0.0f;
    for (int o = 0; o < On; ++o) {
        float l = (mask_first && o == 0) ? -1e9f : row[o];
        float e = __expf(l - mx);
        crow[o] = e;
        sum += e;
    }
    float inv = 1.0f / sum;
    for (int o = 0; o < On; ++o) crow[o] *= inv;
}

// ---------------------------------------------------------------------------
// 5) WMMA s-accumulation:  s[bt,o,d] = sum_i c[bt,i,o] * (W[i,od,:]·xw[bt,i,:] + bias[i,od])
//    One wave per (16-position tile, 16-wide od tile). Per window i:
//      U(16x16) = bias + A(16x8) x B(8x16) via two chained v_wmma_f32_16x16x4_f32,
//    then c-weighted accumulate on the C-fragment (N = lane%16 fixed per lane,
//    so o is a per-lane constant; c varies only across the 8 accumulator rows).
// ---------------------------------------------------------------------------
template<int In, int On, int Od>
__global__ __launch_bounds__(32)
void s_wmma_kernel(const float* __restrict__ xw, const float* __restrict__ Wm,
                   const float* __restrict__ bias, const float* __restrict__ c_buf,
                   float* __restrict__ s_buf) {
    constexpr int OnOd = On * Od;
    constexpr int TO   = 16 / Od;            // output caps spanned by one od-tile
    const int bt0  = blockIdx.x * 16;
    const int ot   = blockIdx.y;             // od tile index
    const int lane = threadIdx.x;
    const int half = lane >> 4;
    const int n    = lane & 15;              // A row (m) and B/C column (n)
    const int o0   = (ot * 16) / Od;

    __shared__ float c_lds[In * 16 * TO];    // [i][m][local o]
    for (int idx = lane; idx < In * 16 * TO; idx += 32) {
        int i   = idx / (16 * TO);
        int rem = idx % (16 * TO);
        int m   = rem / TO;
        int j   = rem % TO;
        c_lds[idx] = c_buf[(size_t)(bt0 + m) * In * On + (size_t)i * On + o0 + j];
    }
    __syncthreads();

    const int jn = n / Od;                   // lane's local o within the tile
    const float* xrow = xw + (size_t)(bt0 + n) * (In * DIN);
    v8f s_acc = {};

    for (int i = 0; i < In; ++i) {
        float bval = bias[(size_t)i * OnOd + ot * 16 + n];
        v8f u = { bval, bval, bval, bval, bval, bval, bval, bval };
        const float* wrow = Wm + ((size_t)i * OnOd + ot * 16 + n) * DIN;
        // A fragment: VGPR0 = A[m][kc+2*half], VGPR1 = A[m][kc+2*half+1]
        v2f a0 = *(const v2f*)(xrow + i * DIN + 2 * half);
        v2f a1 = *(const v2f*)(xrow + i * DIN + 4 + 2 * half);
        // B fragment: VGPR0 = B[kc+2*half][n], VGPR1 = B[kc+2*half+1][n] (K=e contiguous)
        v2f b0 = *(const v2f*)(wrow + 2 * half);
        v2f b1 = *(const v2f*)(wrow + 4 + 2 * half);
        u = __builtin_amdgcn_wmma_f32_16x16x4_f32(false, a0, false, b0, (short)0, u, false, false);
        u = __builtin_amdgcn_wmma_f32_16x16x4_f32(false, a1, false, b1, (short)0, u, false, false);
        #pragma unroll
        for (int r = 0; r < 8; ++r) {
            int m_c = r + 8 * half;          // C-fragment row
            s_acc[r] += c_lds[(i * 16 + m_c) * TO + jn] * u[r];
        }
    }
    #pragma unroll
    for (int r = 0; r < 8; ++r) {
        int m_c = r + 8 * half;
        s_buf[(size_t)(bt0 + m_c) * OnOd + ot * 16 + n] = s_acc[r];
    }
}

// ---------------------------------------------------------------------------
// 6) squash of s over capsule dim -> v
// ---------------------------------------------------------------------------
__global__ void squash_kernel(const float* __restrict__ s, float* __restrict__ v,
                              int Od, int total) {
    int idx = blockIdx.x * blockDim.x + threadIdx.x;   // bt*On + o
    if (idx >= total) return;
    const float* row = s + (size_t)idx * Od;
    float sq = 0.0f;
    for (int d = 0; d < Od; ++d) sq += row[d] * row[d];
    float scale = (sq / (1.0f + sq)) * rsqrtf(sq + 1e-7f);
    float* vrow = v + (size_t)idx * Od;
    for (int d = 0; d < Od; ++d) vrow[d] = scale * row[d];
}

// ---------------------------------------------------------------------------
// 7) WMMA logit update: blog[bt,i,o] += sum_d u_hat[bt,i,o,d] * v[bt,o,d]
//    u_hat tile recomputed via the same 2-WMMA primitive; the d-reduction is a
//    <=16-lane butterfly that never crosses the 16-lane C-fragment halves.
//    Each (bt,i,o) is owned by exactly one block -> plain read-modify-write.
// ---------------------------------------------------------------------------
template<int In, int On, int Od>
__global__ __launch_bounds__(32)
void blog_update_kernel(const float* __restrict__ xw, const float* __restrict__ Wm,
                        const float* __restrict__ bias, const float* __restrict__ v_buf,
                        float* __restrict__ blog) {
    constexpr int OnOd = On * Od;
    const int bt0  = blockIdx.x * 16;
    const int ot   = blockIdx.y;
    const int lane = threadIdx.x;
    const int half = lane >> 4;
    const int n    = lane & 15;
    const int o0   = (ot * 16) / Od;

    __shared__ float v_lds[16 * 16];         // v tile [m][n]
    for (int idx = lane; idx < 256; idx += 32) {
        int m = idx >> 4, nn = idx & 15;
        v_lds[idx] = v_buf[(size_t)(bt0 + m) * OnOd + ot * 16 + nn];
    }
    __syncthreads();

    float vv[8];
    #pragma unroll
    for (int r = 0; r < 8; ++r) vv[r] = v_lds[(r + 8 * half) * 16 + n];

    const float* xrow = xw + (size_t)(bt0 + n) * (In * DIN);
    for (int i = 0; i < In; ++i) {
        float bval = bias[(size_t)i * OnOd + ot * 16 + n];
        v8f u = { bval, bval, bval, bval, bval, bval, bval, bval };
        const float* wrow = Wm + ((size_t)i * OnOd + ot * 16 + n) * DIN;
        v2f a0 = *(const v2f*)(xrow + i * DIN + 2 * half);
        v2f a1 = *(const v2f*)(xrow + i * DIN + 4 + 2 * half);
        v2f b0 = *(const v2f*)(wrow + 2 * half);
        v2f b1 = *(const v2f*)(wrow + 4 + 2 * half);
        u = __builtin_amdgcn_wmma_f32_16x16x4_f32(false, a0, false, b0, (short)0, u, false, false);
        u = __builtin_amdgcn_wmma_f32_16x16x4_f32(false, a1, false, b1, (short)0, u, false, false);
        #pragma unroll
        for (int r = 0; r < 8; ++r) {
            float x = u[r] * vv[r];
            #pragma unroll
            for (int mask = 1; mask < Od; mask <<= 1)
                x += __shfl_xor(x, mask, 32);        // stays within 16-lane half
            if ((n % Od) == 0) {
                int o = o0 + n / Od;
                size_t bidx = (size_t)(bt0 + r + 8 * half) * In * On + (size_t)i * On + o;
                blog[bidx] += x;
            }
        }
    }
}

// ---------------------------------------------------------------------------
// 8) LayerNorm over F features (F = 256 or 1024), grid BT x 256
// ---------------------------------------------------------------------------
template<int F>
__global__ __launch_bounds__(256)
void ln_kernel(const float* __restrict__ src, const float* __restrict__ g,
               const float* __restrict__ b, float* __restrict__ dst) {
    const int bt = blockIdx.x;
    const int t  = threadIdx.x;
    constexpr int PER = F / 256;
    float vals[PER];
    float s = 0.0f;
    #pragma unroll
    for (int p = 0; p < PER; ++p) {
        vals[p] = src[(size_t)bt * F + p * 256 + t];
        s += vals[p];
    }
    __shared__ float red[256];
    float mean = block_reduce_sum(s, red, t, 256) * (1.0f / (float)F);
    float s2 = 0.0f;
    #pragma unroll
    for (int p = 0; p < PER; ++p) {
        float d = vals[p] - mean;
        s2 += d * d;
    }
    float var = block_reduce_sum(s2, red, t, 256) * (1.0f / (float)F);
    float inv = rsqrtf(var + 1e-3f);
    #pragma unroll
    for (int p = 0; p < PER; ++p)
        dst[(size_t)bt * F + p * 256 + t] =
            g[p * 256 + t] * (vals[p] - mean) * inv + b[p * 256 + t];
}

// ---------------------------------------------------------------------------
// 9) capsule lengths + output LayerNorm over 64 class caps
// ---------------------------------------------------------------------------
__global__ __launch_bounds__(64)
void final_kernel(const float* __restrict__ vln, const float* __restrict__ g,
                  const float* __restrict__ b, float* __restrict__ out) {
    const int bt = blockIdx.x;
    const int o  = threadIdx.x;          // 64 class caps
    const float* row = vln + (size_t)bt * 1024 + o * 16;
    float s = 0.0f;
    #pragma unroll
    for (int d = 0; d < 16; ++d) s += row[d] * row[d];
    float len = sqrtf(s + 1e-7f);
    __shared__ float red[64];
    float mean = block_reduce_sum(len, red, o, 64) * (1.0f / 64.0f);
    float d = len - mean;
    float var = block_reduce_sum(d * d, red, o, 64) * (1.0f / 64.0f);
    out[(size_t)bt * 64 + o] = g[o] * d * rsqrtf(var + 1e-3f) + b[o];
}

// ---------------------------------------------------------------------------
// launcher
// ---------------------------------------------------------------------------
extern "C" void kernel_launch(void* const* d_in, const int* in_sizes, int n_in,
                              void* d_out, int out_size, void* d_ws, size_t ws_size,
                              hipStream_t stream) {
    const float* emb   = (const float*)d_in[0];
    const float* W0    = (const float*)d_in[1];
    const float* b0    = (const float*)d_in[2];
    const float* W1    = (const float*)d_in[3];
    const float* b1    = (const float*)d_in[4];
    const float* gi_g  = (const float*)d_in[5];
    const float* gi_b  = (const float*)d_in[6];
    const float* gm0_g = (const float*)d_in[7];
    const float* gm0_b = (const float*)d_in[8];
    const float* gm1_g = (const float*)d_in[9];
    const float* gm1_b = (const float*)d_in[10];
    const float* go_g  = (const float*)d_in[11];
    const float* go_b  = (const float*)d_in[12];
    float* out = (float*)d_out;

    float* ws  = (float*)d_ws;
    float* x0  = ws;                      // BT*256
    float* xw  = x0  + (size_t)BT * 256;  // BT*160*8
    float* cb  = xw  + (size_t)BT * IN_CAPS * DIN;   // BT*160*64 (max)
    float* bl  = cb  + (size_t)BT * IN_CAPS * 64;    // BT*160*64 (max)
    float* sb  = bl  + (size_t)BT * IN_CAPS * 64;    // BT*1024 (max)
    float* vb  = sb  + (size_t)BT * 1024;            // BT*1024
    float* vln = vb  + (size_t)BT * 1024;            // BT*1024

    const dim3 grid0(BT / 16, 256 / 16);   // layer0: 16 od tiles
    const dim3 grid1(BT / 16, 1024 / 16);  // layer1: 64 od tiles

    preprocess_kernel<<<BT, 256, 0, stream>>>(emb, gi_g, gi_b, x0);

    // ---- routing layer 0: 160 -> 32 caps x 8 ----
    gather_kernel<<<BT, 256, 0, stream>>>(x0, xw);
    {
        int cnt = BT * IN_CAPS * 32;
        zero_kernel<<<(cnt + 255) / 256, 256, 0, stream>>>(bl, cnt);
    }
    for (int it = 0; it < 3; ++it) {
        softmax_kernel<<<(BT * IN_CAPS + 255) / 256, 256, 0, stream>>>(bl, cb, 32, 0, BT * IN_CAPS);
        s_wmma_kernel<IN_CAPS, 32, 8><<<grid0, 32, 0, stream>>>(xw, W0, b0, cb, sb);
        squash_kernel<<<(BT * 32 + 255) / 256, 256, 0, stream>>>(sb, vb, 8, BT * 32);
        if (it < 2)
            blog_update_kernel<IN_CAPS, 32, 8><<<grid0, 32, 0, stream>>>(xw, W0, b0, vb, bl);
    }
    ln_kernel<256><<<BT, 256, 0, stream>>>(vb, gm0_g, gm0_b, vln);

    // ---- routing layer 1: 160 -> 64 caps x 16 (first cap masked) ----
    gather_kernel<<<BT, 256, 0, stream>>>(vln, xw);
    {
        int cnt = BT * IN_CAPS * 64;
        zero_kernel<<<(cnt + 255) / 256, 256, 0, stream>>>(bl, cnt);
    }
    for (int it = 0; it < 3; ++it) {
        softmax_kernel<<<(BT * IN_CAPS + 255) / 256, 256, 0, stream>>>(bl, cb, 64, 1, BT * IN_CAPS);
        s_wmma_kernel<IN_CAPS, 64, 16><<<grid1, 32, 0, stream>>>(xw, W1, b1, cb, sb);
        squash_kernel<<<(BT * 64 + 255) / 256, 256, 0, stream>>>(sb, vb, 16, BT * 64);
        if (it < 2)
            blog_update_kernel<IN_CAPS, 64, 16><<<grid1, 32, 0, stream>>>(xw, W1, b1, vb, bl);
    }
    ln_kernel<1024><<<BT, 256, 0, stream>>>(vb, gm1_g, gm1_b, vln);

    final_kernel<<<BT, 64, 0, stream>>>(vln, go_g, go_b, out);
}